// GATEncoder_28467043238275
// MI455X (gfx1250) — compile-verified
//
#include <hip/hip_runtime.h>
#include <math.h>

typedef __attribute__((ext_vector_type(2))) float v2f;
typedef __attribute__((ext_vector_type(8))) float v8f;

#define K_DIM 128          // inner dim of both GEMMs (IN and HEADS*HID)
#define NEG_SLOPE 0.2f

// ---------------------------------------------------------------------------
// fill: raw 32-bit pattern fill (used for 0 and -inf init)
// ---------------------------------------------------------------------------
__global__ void fill_u32_kernel(unsigned* __restrict__ p, int n, unsigned v) {
    int i = blockIdx.x * blockDim.x + threadIdx.x;
    if (i < n) p[i] = v;
}

// ---------------------------------------------------------------------------
// WMMA GEMM: C[M,NCOL] = A[M,128] * B[128,NCOL]   (row-major, f32)
// B staged in LDS. One 16x16 output tile per wave, 4 waves/block.
// V_WMMA_F32_16X16X4_F32, K advanced 4 per step (32 WMMAs per tile).
// A 16x4 layout: lanes 0-15 hold K=k0,k0+1 ; lanes 16-31 hold K=k0+2,k0+3.
// B 4x16 layout: lane (l&15) = column, half selects K pair.
// C 16x16 layout: vgpr i -> row i (lanes 0-15) / row 8+i (lanes 16-31).
// NCOL is a template arg so all addressing folds to shifts/immediates.
// ---------------------------------------------------------------------------
template <int NCOL>
__global__ __launch_bounds__(128)
void wmma_gemm_kernel(const float* __restrict__ A, const float* __restrict__ B,
                      float* __restrict__ C, int M, int mtiles) {
    __shared__ float sW[K_DIM * NCOL];

    for (int i = threadIdx.x * 4; i < K_DIM * NCOL; i += 128 * 4)
        *(float4*)&sW[i] = *(const float4*)&B[i];
    __syncthreads();

    const int wave = threadIdx.x >> 5;
    const int lane = threadIdx.x & 31;
    const int tile = blockIdx.x * 4 + wave;
    constexpr int NTILES = NCOL / 16;
    if (tile >= mtiles * NTILES) return;          // wave-uniform

    const int mt = tile / NTILES;
    const int nt = tile - mt * NTILES;
    const int row0 = mt * 16;
    const int col0 = nt * 16;
    const int half = lane >> 4;                   // 0: K pair {0,1}, 1: {2,3}
    const int l    = lane & 15;

    int ar = row0 + l;
    if (ar >= M) ar = M - 1;                      // clamp (stores are guarded)
    const float* arow = A + (size_t)ar * K_DIM;
    const float* bcol = sW + col0 + l;

    v8f acc = {};
    #pragma unroll 4
    for (int k0 = 0; k0 < K_DIM; k0 += 4) {
        v2f a = *(const v2f*)(arow + k0 + 2 * half);     // 8B-aligned
        v2f b;
        b.x = bcol[(k0 + 2 * half)     * NCOL];
        b.y = bcol[(k0 + 2 * half + 1) * NCOL];
        acc = __builtin_amdgcn_wmma_f32_16x16x4_f32(
                  false, a, false, b, (short)0, acc, false, false);
    }

    float* cp = C + (size_t)(row0 + half * 8) * NCOL + col0 + l;
    if (row0 + 16 <= M) {                         // full tile: unguarded stores
        #pragma unroll
        for (int i = 0; i < 8; ++i) cp[i * NCOL] = acc[i];
    } else {                                      // ragged tail (unused when M%16==0)
        #pragma unroll
        for (int i = 0; i < 8; ++i)
            if (row0 + half * 8 + i < M) cp[i * NCOL] = acc[i];
    }
}

// ---------------------------------------------------------------------------
// per-(node,head) attention coefficients: a_s = <h, att_src>, a_d = <h, att_dst>
// ---------------------------------------------------------------------------
__global__ void att_coeff_kernel(const float* __restrict__ h,
                                 const float* __restrict__ att_s,
                                 const float* __restrict__ att_d,
                                 float* __restrict__ a_s, float* __restrict__ a_d,
                                 int N, int H, int CH) {
    int i = blockIdx.x * blockDim.x + threadIdx.x;
    if (i >= N * H) return;
    int n  = i / H;
    int hh = i - n * H;
    const float* hp = h + (size_t)n * H * CH + (size_t)hh * CH;
    const float* sp = att_s + hh * CH;
    const float* dp = att_d + hh * CH;
    float ss = 0.f, dd = 0.f;
    for (int c = 0; c < CH; ++c) { ss += hp[c] * sp[c]; dd += hp[c] * dp[c]; }
    a_s[i] = ss;
    a_d[i] = dd;
}

// order-preserving float atomic max via integer atomics
__device__ inline void atomicMaxF(float* addr, float v) {
    int iv = __float_as_int(v);
    if (iv >= 0) atomicMax((int*)addr, iv);
    else         atomicMin((unsigned int*)addr, (unsigned int)iv);
}

// ---------------------------------------------------------------------------
// edge pass 1: e = leakyrelu(a_s[src]+a_d[dst]); segment-max into emax[dst]
// self loops synthesized for e >= E
// ---------------------------------------------------------------------------
__global__ void edge_max_kernel(const int* __restrict__ ei, int E, int ET,
                                const float* __restrict__ a_s,
                                const float* __restrict__ a_d,
                                float* __restrict__ ebuf, float* __restrict__ emax,
                                int H) {
    int e = blockIdx.x * blockDim.x + threadIdx.x;
    if (e >= ET) return;
    int src = (e < E) ? ei[e]     : (e - E);
    int dst = (e < E) ? ei[E + e] : (e - E);
    for (int h = 0; h < H; ++h) {
        float v = a_s[src * H + h] + a_d[dst * H + h];
        v = (v > 0.f) ? v : NEG_SLOPE * v;
        ebuf[(size_t)e * H + h] = v;
        atomicMaxF(&emax[dst * H + h], v);
    }
}

// ---------------------------------------------------------------------------
// edge pass 2: ex = exp(e - emax[dst]); denom[dst] += ex  (ebuf updated inplace)
// ---------------------------------------------------------------------------
__global__ void edge_expsum_kernel(const int* __restrict__ ei, int E, int ET,
                                   const float* __restrict__ emax,
                                   float* __restrict__ ebuf,
                                   float* __restrict__ denom, int H) {
    int e = blockIdx.x * blockDim.x + threadIdx.x;
    if (e >= ET) return;
    int dst = (e < E) ? ei[E + e] : (e - E);
    for (int h = 0; h < H; ++h) {
        float ex = __expf(ebuf[(size_t)e * H + h] - emax[dst * H + h]);
        ebuf[(size_t)e * H + h] = ex;
        atomicAdd(&denom[dst * H + h], ex);
    }
}

// ---------------------------------------------------------------------------
// edge pass 3: agg[dst, :] += h[src, :] * (ex / denom[dst])
// one block per edge, one thread per channel (blockDim = H*CH)
// ---------------------------------------------------------------------------
__global__ void edge_agg_kernel(const int* __restrict__ ei, int E,
                                const float* __restrict__ hfeat,
                                const float* __restrict__ ebuf,
                                const float* __restrict__ denom,
                                float* __restrict__ agg, int H, int CH) {
    int e = blockIdx.x;
    int c = threadIdx.x;                 // 0 .. H*CH-1
    int h = c / CH;
    int src = (e < E) ? ei[e]     : (e - E);
    int dst = (e < E) ? ei[E + e] : (e - E);
    float alpha = ebuf[(size_t)e * H + h] / denom[dst * H + h];
    float val   = hfeat[(size_t)src * H * CH + c] * alpha;
    atomicAdd(&agg[(size_t)dst * H * CH + c], val);
}

// ---------------------------------------------------------------------------
// elementwise: a = elu(a + b[broadcast])  (in place, layer-1 epilogue)
// ---------------------------------------------------------------------------
__global__ void bias_elu_kernel(float* __restrict__ a, const float* __restrict__ b,
                                int total, int F) {
    int i = blockIdx.x * blockDim.x + threadIdx.x;
    if (i >= total) return;
    float v = a[i] + b[i % F];
    a[i] = (v > 0.f) ? v : (__expf(v) - 1.f);
}

// elementwise: a += b[broadcast]  (layer-2 epilogue into d_out)
__global__ void bias_add_kernel(float* __restrict__ a, const float* __restrict__ b,
                                int total, int F) {
    int i = blockIdx.x * blockDim.x + threadIdx.x;
    if (i >= total) return;
    a[i] += b[i % F];
}

// ---------------------------------------------------------------------------
extern "C" void kernel_launch(void* const* d_in, const int* in_sizes, int n_in,
                              void* d_out, int out_size, void* d_ws, size_t ws_size,
                              hipStream_t stream) {
    const float* x   = (const float*)d_in[0];
    const int*   ei  = (const int*)  d_in[1];
    const float* W1  = (const float*)d_in[2];
    const float* as1 = (const float*)d_in[3];
    const float* ad1 = (const float*)d_in[4];
    const float* b1  = (const float*)d_in[5];
    const float* W2  = (const float*)d_in[6];
    const float* as2 = (const float*)d_in[7];
    const float* ad2 = (const float*)d_in[8];
    const float* b2  = (const float*)d_in[9];
    float* out = (float*)d_out;

    const int N  = in_sizes[0] / 128;   // nodes
    const int E  = in_sizes[1] / 2;     // raw edges
    const int ET = E + N;               // + self loops
    const int H1 = 4, C1 = 32, F1 = 128;
    const int H2 = 1, C2 = 64, F2 = 64;

    // workspace carve (256B aligned)
    char* w = (char*)d_ws;
    auto carve = [&](size_t bytes) {
        void* p = (void*)w;
        w += (bytes + 255) & ~(size_t)255;
        return p;
    };
    float* h1    = (float*)carve((size_t)N * F1 * 4);
    float* a1s   = (float*)carve((size_t)N * H1 * 4);
    float* a1d   = (float*)carve((size_t)N * H1 * 4);
    float* emax1 = (float*)carve((size_t)N * H1 * 4);
    float* den1  = (float*)carve((size_t)N * H1 * 4);
    float* ebuf1 = (float*)carve((size_t)ET * H1 * 4);
    float* agg1  = (float*)carve((size_t)N * F1 * 4);   // becomes h2-input after ELU
    float* h2    = (float*)carve((size_t)N * F2 * 4);
    float* a2s   = (float*)carve((size_t)N * 4);
    float* a2d   = (float*)carve((size_t)N * 4);
    float* emax2 = (float*)carve((size_t)N * 4);
    float* den2  = (float*)carve((size_t)N * 4);
    float* ebuf2 = (float*)carve((size_t)ET * 4);

    const unsigned NEG_INF = 0xff800000u;
    auto fill = [&](void* p, int n, unsigned v) {
        fill_u32_kernel<<<(n + 255) / 256, 256, 0, stream>>>((unsigned*)p, n, v);
    };
    fill(emax1, N * H1, NEG_INF);
    fill(den1,  N * H1, 0u);
    fill(agg1,  N * F1, 0u);
    fill(emax2, N,      NEG_INF);
    fill(den2,  N,      0u);
    fill(out,   N * F2, 0u);

    const int eb = 256, eg = (ET + eb - 1) / eb;

    // ---------------- layer 1: GATConv(128 -> 32, heads=4, concat) ----------
    {
        int mt = (N + 15) / 16;
        int blocks = (mt * (F1 / 16) + 3) / 4;
        wmma_gemm_kernel<128><<<blocks, 128, 0, stream>>>(x, W1, h1, N, mt);
    }
    att_coeff_kernel<<<(N * H1 + 255) / 256, 256, 0, stream>>>(
        h1, as1, ad1, a1s, a1d, N, H1, C1);
    edge_max_kernel<<<eg, eb, 0, stream>>>(ei, E, ET, a1s, a1d, ebuf1, emax1, H1);
    edge_expsum_kernel<<<eg, eb, 0, stream>>>(ei, E, ET, emax1, ebuf1, den1, H1);
    edge_agg_kernel<<<ET, F1, 0, stream>>>(ei, E, h1, ebuf1, den1, agg1, H1, C1);
    bias_elu_kernel<<<(N * F1 + 255) / 256, 256, 0, stream>>>(agg1, b1, N * F1, F1);

    // ---------------- layer 2: GATConv(128 -> 64, heads=1) ------------------
    {
        int mt = (N + 15) / 16;
        int blocks = (mt * (F2 / 16) + 3) / 4;
        wmma_gemm_kernel<64><<<blocks, 128, 0, stream>>>(agg1, W2, h2, N, mt);
    }
    att_coeff_kernel<<<(N + 255) / 256, 256, 0, stream>>>(
        h2, as2, ad2, a2s, a2d, N, H2, C2);
    edge_max_kernel<<<eg, eb, 0, stream>>>(ei, E, ET, a2s, a2d, ebuf2, emax2, H2);
    edge_expsum_kernel<<<eg, eb, 0, stream>>>(ei, E, ET, emax2, ebuf2, den2, H2);
    edge_agg_kernel<<<ET, F2, 0, stream>>>(ei, E, h2, ebuf2, den2, out, H2, C2);
    bias_add_kernel<<<(N * F2 + 255) / 256, 256, 0, stream>>>(out, b2, N * F2, F2);
}